// MaskedCrossentropyLoss_82489141887384
// MI455X (gfx1250) — compile-verified
//
#include <hip/hip_runtime.h>
#include <stdint.h>

// Problem geometry (fixed by the reference)
#define BS      16
#define SEQ     512
#define VOCAB   32000
#define NM      80
#define NPAIR   (BS * NM)          // 1280
#define V4      (VOCAB / 4)        // 8000 float4 per row
#define THREADS 256
#define NSTAGE  ((V4 + THREADS - 1) / THREADS)   // 32 stages (last partial)
#define NBUF    4                                // async ring depth

typedef float f4 __attribute__((ext_vector_type(4)));
// Exact vector type the async-LDS builtin expects (from hipcc diagnostic):
typedef int v4i __attribute__((__vector_size__(16)));
typedef __attribute__((address_space(1))) v4i gv4i;   // global
typedef __attribute__((address_space(3))) v4i lv4i;   // LDS

// ---- gfx1250 async global->LDS path (guarded so we always compile) ----
#if defined(__has_builtin)
#  if __has_builtin(__builtin_amdgcn_global_load_async_to_lds_b128)
#    define HAVE_ASYNC_LDS 1
#  endif
#  if __has_builtin(__builtin_amdgcn_s_wait_asynccnt)
#    define HAVE_WAIT_ASYNC 1
#  endif
#endif

template <int N>
__device__ static inline void wait_async() {
#if defined(HAVE_WAIT_ASYNC)
  __builtin_amdgcn_s_wait_asynccnt(N);
#else
  if constexpr (N == 0)      asm volatile("s_wait_asynccnt 0x0" ::: "memory");
  else if constexpr (N == 1) asm volatile("s_wait_asynccnt 0x1" ::: "memory");
  else if constexpr (N == 2) asm volatile("s_wait_asynccnt 0x2" ::: "memory");
  else                       asm volatile("s_wait_asynccnt 0x3" ::: "memory");
#endif
  asm volatile("" ::: "memory");
}

// ---------------------------------------------------------------------------
// Kernel: per (row, masked-pos) pair: argmax over VOCAB + gather at target.
// One block per pair; row streamed through LDS with a 4-deep async DMA ring
// (up to 3 transfers in flight per wave; each thread reads back only the slot
// it DMA'd, so per-wave ASYNCcnt ordering suffices and no barrier is needed).
// ---------------------------------------------------------------------------
__global__ __launch_bounds__(THREADS) void argmax_gather_kernel(
    const float* __restrict__ outp, const int* __restrict__ target,
    const int* __restrict__ midx, float* __restrict__ logp_o,
    int* __restrict__ corr_o) {
#if defined(HAVE_ASYNC_LDS)
  __shared__ f4 stage[NBUF][THREADS];   // 16 KB ring
#endif
  __shared__ float swv[THREADS / 32];
  __shared__ int swi[THREADS / 32];

  const int pair = blockIdx.x;
  const int b = pair / NM;
  const int pos = midx[pair];
  const float* row = outp + ((size_t)(b * SEQ + pos)) * VOCAB;
  const f4* rowv = (const f4*)row;
  const int t = threadIdx.x;

  // Hoist the NLL gather so its latency overlaps the streaming loop.
  int tg = 0;
  float lp = 0.f;
  if (t == 0) {
    tg = target[pair];
    lp = row[tg];
  }

  float bv = -__builtin_inff();
  int bi = 0x7fffffff;

#if defined(HAVE_ASYNC_LDS)
  // Prologue: fill 3 of the 4 ring slots (indices all < V4: 768 < 8000).
  __builtin_amdgcn_global_load_async_to_lds_b128(
      (gv4i*)(rowv + 0 * THREADS + t), (lv4i*)&stage[0][t], 0, 0);
  __builtin_amdgcn_global_load_async_to_lds_b128(
      (gv4i*)(rowv + 1 * THREADS + t), (lv4i*)&stage[1][t], 0, 0);
  __builtin_amdgcn_global_load_async_to_lds_b128(
      (gv4i*)(rowv + 2 * THREADS + t), (lv4i*)&stage[2][t], 0, 0);

  for (int s = 0; s < NSTAGE; ++s) {
    if (s + 3 < NSTAGE) {
      // Clamp keeps every wave issuing uniformly -> ASYNCcnt stays in lockstep.
      int nidx = (s + 3) * THREADS + t;
      if (nidx > V4 - 1) nidx = V4 - 1;
      __builtin_amdgcn_global_load_async_to_lds_b128(
          (gv4i*)(rowv + nidx), (lv4i*)&stage[(s + 3) & (NBUF - 1)][t], 0, 0);
      wait_async<3>();          // stage s complete, 3 still in flight
    } else if (s + 2 < NSTAGE) {
      wait_async<2>();          // draining
    } else if (s + 1 < NSTAGE) {
      wait_async<1>();
    } else {
      wait_async<0>();
    }
    int cidx = s * THREADS + t;
    if (cidx < V4) {
      __attribute__((address_space(3))) const f4* p =
          (__attribute__((address_space(3))) const f4*)&stage[s & (NBUF - 1)][t];
      asm volatile("" : "+v"(p)::"memory");  // opacify: LDS written by DMA engine
      f4 v = *p;
      int base = cidx * 4;
      if (v.x > bv) { bv = v.x; bi = base; }
      if (v.y > bv) { bv = v.y; bi = base + 1; }
      if (v.z > bv) { bv = v.z; bi = base + 2; }
      if (v.w > bv) { bv = v.w; bi = base + 3; }
    }
  }
#else
  // Fallback: direct vectorized global loads.
  for (int i = t; i < V4; i += THREADS) {
    f4 v = rowv[i];
    int base = i * 4;
    if (v.x > bv) { bv = v.x; bi = base; }
    if (v.y > bv) { bv = v.y; bi = base + 1; }
    if (v.z > bv) { bv = v.z; bi = base + 2; }
    if (v.w > bv) { bv = v.w; bi = base + 3; }
  }
#endif

  // wave32 reduction (argmax, lowest-index tie-break)
  for (int off = 16; off > 0; off >>= 1) {
    float ov = __shfl_xor(bv, off, 32);
    int oi = __shfl_xor(bi, off, 32);
    if (ov > bv || (ov == bv && oi < bi)) { bv = ov; bi = oi; }
  }
  int w = t >> 5;
  if ((t & 31) == 0) { swv[w] = bv; swi[w] = bi; }
  __syncthreads();
  if (t == 0) {
    float fv = swv[0];
    int fi = swi[0];
    for (int k = 1; k < THREADS / 32; ++k) {
      if (swv[k] > fv || (swv[k] == fv && swi[k] < fi)) { fv = swv[k]; fi = swi[k]; }
    }
    logp_o[pair] = lp;                  // gather for NLL (hoisted load)
    corr_o[pair] = (fi == tg) ? 1 : 0;  // argmax hit
  }
}

// ---------------------------------------------------------------------------
// Kernel: recover masked positions per row (stable order == first-80 True).
// ---------------------------------------------------------------------------
__global__ void build_idx_kernel(const unsigned char* __restrict__ mask,
                                 int* __restrict__ midx) {
  int b = threadIdx.x;
  if (b >= BS) return;
  const unsigned char* m = mask + (size_t)b * SEQ;
  int k = 0;
  for (int s = 0; s < SEQ && k < NM; ++s) {
    if (m[s]) midx[b * NM + k++] = s;
  }
}

// ---------------------------------------------------------------------------
// Kernel: deterministic fixed-order reduction of the 1280 partials.
// ---------------------------------------------------------------------------
__global__ __launch_bounds__(256) void finalize_kernel(
    const float* __restrict__ logp, const int* __restrict__ corr,
    float* __restrict__ out) {
  __shared__ float sf[256];
  __shared__ int si[256];
  int t = threadIdx.x;
  float s = 0.f;
  int c = 0;
  for (int i = t; i < NPAIR; i += 256) { s += logp[i]; c += corr[i]; }
  sf[t] = s; si[t] = c;
  __syncthreads();
  for (int off = 128; off > 0; off >>= 1) {
    if (t < off) { sf[t] += sf[t + off]; si[t] += si[t + off]; }
    __syncthreads();
  }
  if (t == 0) {
    out[0] = -sf[0] / (float)NPAIR;        // sum_b(-mean_j logp)/BS
    out[1] = (float)si[0] / (float)NPAIR;  // accuracy
  }
}

// ---------------------------------------------------------------------------
extern "C" void kernel_launch(void* const* d_in, const int* in_sizes, int n_in,
                              void* d_out, int out_size, void* d_ws,
                              size_t ws_size, hipStream_t stream) {
  const float* output = (const float*)d_in[0];          // (16,512,32000) f32
  const int* target = (const int*)d_in[1];              // (16,80) int
  const unsigned char* mask = (const unsigned char*)d_in[2];  // (16,512) bool

  char* ws = (char*)d_ws;
  int* ws_idx = (int*)ws;                               // 1280 ints
  float* ws_logp = (float*)(ws + NPAIR * sizeof(int));  // 1280 floats
  int* ws_corr = (int*)(ws + NPAIR * (sizeof(int) + sizeof(float)));

  build_idx_kernel<<<1, 32, 0, stream>>>(mask, ws_idx);
  argmax_gather_kernel<<<NPAIR, THREADS, 0, stream>>>(output, target, ws_idx,
                                                      ws_logp, ws_corr);
  finalize_kernel<<<1, 256, 0, stream>>>(ws_logp, ws_corr, (float*)d_out);
}